// PSTTransformer_5411658793002
// MI455X (gfx1250) — compile-verified
//
#include <hip/hip_runtime.h>
#include <hip/hip_bf16.h>
#include <math.h>

typedef float v2f __attribute__((ext_vector_type(2)));
typedef float v8f __attribute__((ext_vector_type(8)));

#define B_    2
#define T_    4
#define N_    1024
#define L_    4
#define MA_   256      // anchors per frame (N/4)
#define KNB_  32       // ball-query K
#define C_    512
#define HEADS_ 8
#define DH_   64
#define MTOK_ 1024     // tokens per batch = L_*MA_
#define ROWS_ 2048     // B_*MTOK_
#define MLP_  2048

__device__ __forceinline__ v8f wmma4(v2f a, v2f b, v8f c) {
  // D = A(16x4 f32) * B(4x16 f32) + C(16x16 f32)
  return __builtin_amdgcn_wmma_f32_16x16x4_f32(false, a, false, b, (short)0, c,
                                               false, false);
}

__device__ __forceinline__ float gelu_exact(float x) {
  return 0.5f * x * (1.0f + erff(x * 0.7071067811865475f));
}

// ---------------------------------------------------------------------------
// Farthest point sampling: one block per (b, frame), 1024 threads.
// Writes anchor coordinates (B*L, 256, 3).
// ---------------------------------------------------------------------------
__global__ void fps_kernel(const float* __restrict__ xyzs,
                           float* __restrict__ anchors) {
  __shared__ float sval[N_];
  __shared__ int   sidx[N_];
  __shared__ float last[3];
  const int blk = blockIdx.x;          // b*L_ + l  (anchor frame = original l)
  const int tid = threadIdx.x;
  const float* frame = xyzs + (size_t)blk * N_ * 3;
  const float px = frame[tid * 3 + 0];
  const float py = frame[tid * 3 + 1];
  const float pz = frame[tid * 3 + 2];
  float dist = 1e10f;
  float* aout = anchors + (size_t)blk * MA_ * 3;
  if (tid == 0) {
    last[0] = px; last[1] = py; last[2] = pz;   // point 0 is first anchor
    aout[0] = px; aout[1] = py; aout[2] = pz;
  }
  __syncthreads();
  for (int it = 1; it < MA_; ++it) {
    const float dx = px - last[0], dy = py - last[1], dz = pz - last[2];
    dist = fminf(dist, dx * dx + dy * dy + dz * dz);
    sval[tid] = dist; sidx[tid] = tid;
    __syncthreads();
    for (int s = 512; s > 0; s >>= 1) {
      if (tid < s) {
        const float v2v = sval[tid + s];
        const int   i2  = sidx[tid + s];
        if (v2v > sval[tid] || (v2v == sval[tid] && i2 < sidx[tid])) {
          sval[tid] = v2v; sidx[tid] = i2;     // argmax, first-index tiebreak
        }
      }
      __syncthreads();
    }
    const int nxt = sidx[0];
    __syncthreads();
    if (tid == nxt) {
      last[0] = px; last[1] = py; last[2] = pz;
      aout[it * 3 + 0] = px; aout[it * 3 + 1] = py; aout[it * 3 + 2] = pz;
    }
    __syncthreads();
  }
}

// ---------------------------------------------------------------------------
// p4dconv: one block per (b, l, anchor), 1024 threads.
// Ball query (first 32 indices with d2 < 1 in index order) via prefix scan,
// then feats[c] = relu( max over (frame, k) of Wd[c] . [disp, dt] ).
// ---------------------------------------------------------------------------
__global__ void p4dconv_kernel(const float* __restrict__ xyzs,
                               const float* __restrict__ anchors,
                               const float* __restrict__ Wd,
                               float* __restrict__ feats) {
  __shared__ int   sscan[N_];
  __shared__ float spx[KNB_], spy[KNB_], spz[KNB_];
  __shared__ float s_p0[3];
  const int blk = blockIdx.x;                 // ((b*L_+l)*MA_ + m)
  const int m   = blk % MA_;
  const int bl  = blk / MA_;                  // b*L_ + l
  const int l   = bl % L_;
  const int b   = bl / L_;
  const int tid = threadIdx.x;

  const float* arow = anchors + ((size_t)bl * MA_ + m) * 3;
  const float ax = arow[0], ay = arow[1], az = arow[2];

  float w0 = 0.f, w1 = 0.f, w2 = 0.f, w3 = 0.f;
  if (tid < C_) {
    w0 = Wd[tid * 4 + 0]; w1 = Wd[tid * 4 + 1];
    w2 = Wd[tid * 4 + 2]; w3 = Wd[tid * 4 + 3];
  }
  float maxv = -3.4e38f;

  for (int di = -1; di <= 1; ++di) {
    int fi = l + di; fi = fi < 0 ? 0 : (fi > T_ - 1 ? T_ - 1 : fi);
    const float* nb = xyzs + ((size_t)b * T_ + fi) * N_ * 3;
    const float qx = nb[tid * 3 + 0];
    const float qy = nb[tid * 3 + 1];
    const float qz = nb[tid * 3 + 2];
    const float dx = qx - ax, dy = qy - ay, dz = qz - az;
    const int pred = (dx * dx + dy * dy + dz * dz) < 1.0f ? 1 : 0;
    sscan[tid] = pred;
    if (tid == 0) { s_p0[0] = dx; s_p0[1] = dy; s_p0[2] = dz; }
    __syncthreads();
    // Hillis-Steele inclusive scan over 1024 predicates
    for (int off = 1; off < N_; off <<= 1) {
      const int v   = sscan[tid];
      const int add = (tid >= off) ? sscan[tid - off] : 0;
      __syncthreads();
      sscan[tid] = v + add;
      __syncthreads();
    }
    const int incl  = sscan[tid];
    const int total = sscan[N_ - 1];
    const int pos   = incl - pred;
    if (pred && pos < KNB_) { spx[pos] = dx; spy[pos] = dy; spz[pos] = dz; }
    __syncthreads();
    const int cnt = total < KNB_ ? total : KNB_;
    if (tid < KNB_ && tid >= cnt) {
      // pad with first in-ball point; if ball empty, reference falls back to
      // index 0 of the neighbor frame
      spx[tid] = (total > 0) ? spx[0] : s_p0[0];
      spy[tid] = (total > 0) ? spy[0] : s_p0[1];
      spz[tid] = (total > 0) ? spz[0] : s_p0[2];
    }
    __syncthreads();
    if (tid < C_) {
      const float tq = (float)di;
      #pragma unroll 8
      for (int k = 0; k < KNB_; ++k) {
        const float v = w0 * spx[k] + w1 * spy[k] + w2 * spz[k] + w3 * tq;
        maxv = fmaxf(maxv, v);
      }
    }
    __syncthreads();
  }
  if (tid < C_)
    feats[((size_t)bl * MA_ + m) * C_ + tid] = fmaxf(maxv, 0.0f);  // fused relu
}

// ---------------------------------------------------------------------------
// LayerNorm over C=512, one block per token row.
// ---------------------------------------------------------------------------
__global__ void ln_kernel(const float* __restrict__ x, const float* __restrict__ g,
                          const float* __restrict__ bta, float* __restrict__ y) {
  __shared__ float red[256];
  const int row = blockIdx.x, t = threadIdx.x;
  const float* xr = x + (size_t)row * C_;
  const float a0 = xr[t], a1 = xr[t + 256];
  red[t] = a0 + a1; __syncthreads();
  for (int s = 128; s > 0; s >>= 1) { if (t < s) red[t] += red[t + s]; __syncthreads(); }
  const float mean = red[0] * (1.0f / C_);
  __syncthreads();
  const float d0 = a0 - mean, d1 = a1 - mean;
  red[t] = d0 * d0 + d1 * d1; __syncthreads();
  for (int s = 128; s > 0; s >>= 1) { if (t < s) red[t] += red[t + s]; __syncthreads(); }
  const float inv = rsqrtf(red[0] * (1.0f / C_) + 1e-5f);
  float* yr = y + (size_t)row * C_;
  yr[t]       = d0 * inv * g[t]       + bta[t];
  yr[t + 256] = d1 * inv * g[t + 256] + bta[t + 256];
}

// ---------------------------------------------------------------------------
// fp32 WMMA GEMM, 32x32 tile per wave (2x2 register blocking):
//   Cout(MxN) = act(A(MxK) @ W(NxK)^T + bias) [+ resid]
// Per k-step: 2 A-frag + 2 B-frag b64 loads feed 4 WMMAs (1 load : 1 WMMA).
// act: 0 = none, 1 = exact gelu.
// ---------------------------------------------------------------------------
__global__ void gemm_wmma(const float* __restrict__ A, int lda,
                          const float* __restrict__ W, int ldw,
                          const float* __restrict__ bias,
                          const float* __restrict__ resid,
                          float* __restrict__ Cout, int ldc,
                          int Kdim, int act) {
  const int j0 = blockIdx.x * 32, i0 = blockIdx.y * 32;
  const int lane = threadIdx.x & 31;
  const int sub = lane & 15, hi = lane >> 4;
  // A frag: lane sub holds row i0+sub, K = {k0+2*hi, k0+2*hi+1} (float2)
  // B frag: B[k][n] = W[n][k] -> lane sub holds col j0+sub, same K pair
  const float* arow0 = A + (size_t)(i0 + sub) * lda + 2 * hi;
  const float* arow1 = arow0 + (size_t)16 * lda;
  const float* wrow0 = W + (size_t)(j0 + sub) * ldw + 2 * hi;
  const float* wrow1 = wrow0 + (size_t)16 * ldw;
  __builtin_prefetch(arow0, 0, 0);
  __builtin_prefetch(wrow0, 0, 0);
  v8f acc00 = {0.f, 0.f, 0.f, 0.f, 0.f, 0.f, 0.f, 0.f};
  v8f acc01 = acc00, acc10 = acc00, acc11 = acc00;
  #pragma unroll 4
  for (int k0 = 0; k0 < Kdim; k0 += 4) {
    const v2f a0 = *(const v2f*)(arow0 + k0);
    const v2f a1 = *(const v2f*)(arow1 + k0);
    const v2f b0 = *(const v2f*)(wrow0 + k0);
    const v2f b1 = *(const v2f*)(wrow1 + k0);
    acc00 = wmma4(a0, b0, acc00);
    acc01 = wmma4(a0, b1, acc01);
    acc10 = wmma4(a1, b0, acc10);
    acc11 = wmma4(a1, b1, acc11);
  }
  const float bj0 = bias ? bias[j0 + sub]      : 0.0f;
  const float bj1 = bias ? bias[j0 + 16 + sub] : 0.0f;
  #pragma unroll
  for (int r = 0; r < 8; ++r) {
    const int row0 = i0 + r + 8 * hi;    // C layout: row = r + 8*hi, col = sub
    const int row1 = row0 + 16;
    const int col0 = j0 + sub;
    const int col1 = col0 + 16;
    float v00 = acc00[r] + bj0, v01 = acc01[r] + bj1;
    float v10 = acc10[r] + bj0, v11 = acc11[r] + bj1;
    if (act == 1) {
      v00 = gelu_exact(v00); v01 = gelu_exact(v01);
      v10 = gelu_exact(v10); v11 = gelu_exact(v11);
    }
    if (resid) {
      v00 += resid[(size_t)row0 * ldc + col0];
      v01 += resid[(size_t)row0 * ldc + col1];
      v10 += resid[(size_t)row1 * ldc + col0];
      v11 += resid[(size_t)row1 * ldc + col1];
    }
    Cout[(size_t)row0 * ldc + col0] = v00;
    Cout[(size_t)row0 * ldc + col1] = v01;
    Cout[(size_t)row1 * ldc + col0] = v10;
    Cout[(size_t)row1 * ldc + col1] = v11;
  }
}

// ---------------------------------------------------------------------------
// Attention: one block per (b, h, 16-query tile); 256 threads (8 waves).
// Phase 1: S(16x1024) = QK^T/8 via WMMA into LDS (Q fragments preloaded).
// Phase 2: row softmax (unnormalized exp, row sums kept).
// Phase 3: waves 0-3: O = P@V via WMMA (A-fragments from LDS, ds_load_b64);
//          waves 4-7: wx = P@xyz (3 cols). Epilogue folds 1/rowsum and adds
//          Wsp @ (wx/rowsum - xyz_i) spatial displacement.
// ---------------------------------------------------------------------------
__global__ void attn_kernel(const float* __restrict__ qkv,
                            const float* __restrict__ xyzf,
                            const float* __restrict__ Wsp,
                            float* __restrict__ attn_out) {
  extern __shared__ float smem[];
  float* S    = smem;               // 16*1024
  float* red  = S + 16 * 1024;      // 256
  float* rmax = red + 256;          // 16
  float* rsum = rmax + 16;          // 16
  float* wx   = rsum + 16;          // 48
  const int b = blockIdx.z, h = blockIdx.y, i0 = blockIdx.x * 16;
  const int t = threadIdx.x;
  const int lane = t & 31, w = t >> 5;
  const int sub = lane & 15, hi = lane >> 4;

  // ---- Phase 1: scores ----
  const float* qbase =
      qkv + ((size_t)(b * MTOK_ + i0 + sub) * 3 + 0) * C_ + h * DH_ + 2 * hi;
  v2f qa[16];
  #pragma unroll
  for (int ks = 0; ks < 16; ++ks) qa[ks] = *(const v2f*)(qbase + 4 * ks);
  for (int jj = 0; jj < 8; ++jj) {
    const int j0 = (w * 8 + jj) * 16;
    const float* kbase =
        qkv + ((size_t)(b * MTOK_ + j0 + sub) * 3 + 1) * C_ + h * DH_ + 2 * hi;
    v8f acc = {0.f, 0.f, 0.f, 0.f, 0.f, 0.f, 0.f, 0.f};
    #pragma unroll
    for (int ks = 0; ks < 16; ++ks) {
      const v2f kb = *(const v2f*)(kbase + 4 * ks);
      acc = wmma4(qa[ks], kb, acc);
    }
    #pragma unroll
    for (int r = 0; r < 8; ++r)
      S[(r + 8 * hi) * 1024 + j0 + sub] = acc[r] * 0.125f;  // 1/sqrt(64)
  }
  __syncthreads();

  // ---- Phase 2: softmax (exp stored unnormalized) ----
  const int row = t >> 4, seg = t & 15;
  float* sr = S + row * 1024 + seg * 64;
  float lm = -3.4e38f;
  for (int c2 = 0; c2 < 64; ++c2) lm = fmaxf(lm, sr[c2]);
  red[t] = lm; __syncthreads();
  if (t < 16) {
    float mm = -3.4e38f;
    for (int s2 = 0; s2 < 16; ++s2) mm = fmaxf(mm, red[t * 16 + s2]);
    rmax[t] = mm;
  }
  __syncthreads();
  const float rm = rmax[row];
  float ls = 0.0f;
  for (int c2 = 0; c2 < 64; ++c2) {
    const float e = expf(sr[c2] - rm);
    sr[c2] = e; ls += e;
  }
  red[t] = ls; __syncthreads();
  if (t < 16) {
    float ss = 0.0f;
    for (int s2 = 0; s2 < 16; ++s2) ss += red[t * 16 + s2];
    rsum[t] = ss;
  }
  __syncthreads();

  // ---- Phase 3 ----
  v8f acc = {0.f, 0.f, 0.f, 0.f, 0.f, 0.f, 0.f, 0.f};
  const int d0 = (w & 3) * 16;
  if (w < 4) {
    const float* vbase =
        qkv + ((size_t)(b * MTOK_) * 3 + 2) * C_ + h * DH_ + d0 + sub;
    const float* srow = S + sub * 1024 + 2 * hi;   // A-fragment rows from LDS
    for (int ks = 0; ks < 256; ++ks) {
      const v2f a = *(const v2f*)(srow + 4 * ks);  // ds_load_b64
      const int kk = 4 * ks + 2 * hi;
      v2f bb;
      bb.x = vbase[(size_t)kk * 1536];
      bb.y = vbase[(size_t)(kk + 1) * 1536];
      acc = wmma4(a, bb, acc);
    }
  } else {
    const int q = (w - 4) * 32 + lane;             // 0..127, use 48
    if (q < 48) {
      const int rr = q / 3, cc = q % 3;
      const float* srw = S + rr * 1024;
      const float* xb  = xyzf + (size_t)b * MTOK_ * 3 + cc;
      float s = 0.0f;
      for (int j = 0; j < 1024; ++j) s += srw[j] * xb[j * 3];
      wx[q] = s;
    }
  }
  __syncthreads();
  if (w < 4) {
    const int d = d0 + sub;
    const float w0 = Wsp[d * 3 + 0], w1 = Wsp[d * 3 + 1], w2 = Wsp[d * 3 + 2];
    #pragma unroll
    for (int r = 0; r < 8; ++r) {
      const int rr = r + 8 * hi;
      const int i  = i0 + rr;
      const float inv = 1.0f / rsum[rr];
      const float* xi = xyzf + ((size_t)b * MTOK_ + i) * 3;
      const float wd0 = wx[rr * 3 + 0] * inv - xi[0];  // sum(attn)=1
      const float wd1 = wx[rr * 3 + 1] * inv - xi[1];
      const float wd2 = wx[rr * 3 + 2] * inv - xi[2];
      attn_out[((size_t)(b * MTOK_) + i) * C_ + h * DH_ + d] =
          acc[r] * inv + w0 * wd0 + w1 * wd1 + w2 * wd2;
    }
  }
}

// ---------------------------------------------------------------------------
extern "C" void kernel_launch(void* const* d_in, const int* in_sizes, int n_in,
                              void* d_out, int out_size, void* d_ws, size_t ws_size,
                              hipStream_t stream) {
  (void)in_sizes; (void)n_in; (void)out_size; (void)ws_size;
  const float* xyzs = (const float*)d_in[0];
  const float* Wd   = (const float*)d_in[1];
  const float* Wqkv = (const float*)d_in[2];   // (2,1536,512)
  const float* Wsp  = (const float*)d_in[3];   // (2,64,3)
  const float* Wout = (const float*)d_in[4];   // (2,512,512)
  const float* bout = (const float*)d_in[5];
  const float* ln1g = (const float*)d_in[6];
  const float* ln1b = (const float*)d_in[7];
  const float* Wff1 = (const float*)d_in[8];   // (2,2048,512)
  const float* bff1 = (const float*)d_in[9];
  const float* Wff2 = (const float*)d_in[10];  // (2,512,2048)
  const float* bff2 = (const float*)d_in[11];
  const float* ln2g = (const float*)d_in[12];
  const float* ln2b = (const float*)d_in[13];

  float* ws      = (float*)d_ws;
  float* anchors = ws;                          // 6144 floats
  float* feats   = ws + 8192;                   // 2048*512
  float* xln     = feats + (size_t)ROWS_ * C_;  // 2048*512
  float* qkvb    = xln   + (size_t)ROWS_ * C_;  // 2048*1536
  float* attno   = qkvb  + (size_t)ROWS_ * 1536;// 2048*512
  float* ffh     = attno + (size_t)ROWS_ * C_;  // 2048*2048

  fps_kernel<<<B_ * L_, N_, 0, stream>>>(xyzs, anchors);
  p4dconv_kernel<<<B_ * L_ * MA_, N_, 0, stream>>>(xyzs, anchors, Wd, feats);

  const size_t smem_attn = (16 * 1024 + 256 + 16 + 16 + 48) * sizeof(float);
  for (int d = 0; d < 2; ++d) {
    ln_kernel<<<ROWS_, 256, 0, stream>>>(feats, ln1g + d * C_, ln1b + d * C_, xln);
    gemm_wmma<<<dim3(1536 / 32, ROWS_ / 32), 32, 0, stream>>>(
        xln, C_, Wqkv + (size_t)d * 1536 * C_, C_, nullptr, nullptr,
        qkvb, 1536, C_, 0);
    attn_kernel<<<dim3(MTOK_ / 16, HEADS_, B_), 256, smem_attn, stream>>>(
        qkvb, anchors, Wsp + (size_t)d * DH_ * 3, attno);
    gemm_wmma<<<dim3(C_ / 32, ROWS_ / 32), 32, 0, stream>>>(
        attno, C_, Wout + (size_t)d * C_ * C_, C_, bout + d * C_,
        feats, feats, C_, C_, 1);
    ln_kernel<<<ROWS_, 256, 0, stream>>>(feats, ln2g + d * C_, ln2b + d * C_, xln);
    gemm_wmma<<<dim3(MLP_ / 32, ROWS_ / 32), 32, 0, stream>>>(
        xln, C_, Wff1 + (size_t)d * MLP_ * C_, C_, bff1 + d * MLP_,
        nullptr, ffh, MLP_, C_, 1);
    gemm_wmma<<<dim3(C_ / 32, ROWS_ / 32), 32, 0, stream>>>(
        ffh, MLP_, Wff2 + (size_t)d * C_ * MLP_, MLP_, bff2 + d * C_,
        feats, feats, C_, MLP_, 0);
  }
  hipMemcpyAsync(d_out, feats, (size_t)ROWS_ * C_ * sizeof(float),
                 hipMemcpyDeviceToDevice, stream);
}